// Downstream_79182017069223
// MI455X (gfx1250) — compile-verified
//
#include <hip/hip_runtime.h>

// MI455X (gfx1250) fused implementation.
// Single streaming pass over x (256 MiB @ 23.3 TB/s ~= 11.5 us floor) computes
// both the LayerNorm-pool branch and the fp32 WMMA covariance branch.
// Covariance is done with V_WMMA_F32_16X16X4_F32 (fp32-exact WMMA path).
// Chunk staging uses GLOBAL_LOAD_ASYNC_TO_LDS_B128 (ASYNCcnt-tracked DMA into
// LDS, no VGPR roundtrip); falls back to b128+ds_store if builtin is absent.

typedef __attribute__((ext_vector_type(2))) float v2f;
typedef __attribute__((ext_vector_type(8))) float v8f;

#if __has_builtin(__builtin_amdgcn_global_load_async_to_lds_b128)
#define HAVE_ASYNC_LDS 1
typedef int v4i_t __attribute__((vector_size(4 * sizeof(int))));
#define AS3 __attribute__((address_space(3)))
#endif

#define C_CH   64
#define D_DIM  1024
#define N_SP   32          // L / C
#define B_BATCH 32
#define L_TOK  2048
#define TRI    2080
#define K_OUT  40
#define COMB   (D_DIM + TRI)   // 3104
#define KC     256             // K-chunk columns staged in LDS
#define PITCH  260             // 260 mod 64 == 4 -> conflict-free WMMA feed

// ---------------------------------------------------------------------------
// Kernel 1: per-(b,n) slab (64 tokens x 1024 dims).
//  - per-token mean/rstd (branch A stats)
//  - chunked LDS staging feeding (a) pooled partial, (b) WMMA f32 covariance
// ---------------------------------------------------------------------------
__global__ __launch_bounds__(256)
void fused_ln_cov_kernel(const float* __restrict__ x,
                         const float* __restrict__ gamma_pool,
                         const float* __restrict__ beta_pool,
                         float* __restrict__ pooled_part,   // [B*N, D]
                         float* __restrict__ cov_part)      // [B*N, 64*64]
{
    extern __shared__ float lds[];            // xs[64*PITCH] + mrow[64] + rrow[64]
    float* xs   = lds;
    float* mrow = lds + C_CH * PITCH;
    float* rrow = mrow + C_CH;

    const int n    = blockIdx.x;
    const int b    = blockIdx.y;
    const int tid  = threadIdx.x;
    const int lane = tid & 31;
    const int wave = tid >> 5;

    const float* slab = x + (size_t)(b * L_TOK + n * C_CH) * D_DIM;

    // ---- pass 1: per-row mean / rstd (8 rows per wave, coalesced 128B loads)
    for (int rr = 0; rr < 8; ++rr) {
        const int row = wave * 8 + rr;
        const float* rp = slab + (size_t)row * D_DIM;
        float s1 = 0.f, s2 = 0.f;
        for (int i = lane; i < D_DIM; i += 32) {
            float v = rp[i];
            s1 += v; s2 += v * v;
        }
        for (int off = 16; off > 0; off >>= 1) {
            s1 += __shfl_xor(s1, off, 32);
            s2 += __shfl_xor(s2, off, 32);
        }
        if (lane == 0) {
            float m   = s1 * (1.0f / D_DIM);
            float var = s2 * (1.0f / D_DIM) - m * m;
            mrow[row] = m;
            rrow[row] = rsqrtf(var + 1e-5f);
        }
    }
    __syncthreads();

    // ---- WMMA tile assignment: wave w owns 16x16 tiles w and w+8 of 4x4 grid
    // (same tj for both -> compiler shares the B operand between accumulators)
    const int t0  = wave,          t1  = wave + 8;
    const int ti0 = (t0 >> 2) * 16, tj0 = (t0 & 3) * 16;
    const int ti1 = (t1 >> 2) * 16, tj1 = (t1 & 3) * 16;
    const int ln  = lane & 15;     // channel-within-tile (M for A, N for B)
    const int hi  = lane >> 4;     // selects K=+2 half per ISA 16x4 f32 layout

    v8f acc0 = {0.f,0.f,0.f,0.f,0.f,0.f,0.f,0.f};
    v8f acc1 = {0.f,0.f,0.f,0.f,0.f,0.f,0.f,0.f};

    for (int kc = 0; kc < D_DIM / KC; ++kc) {
#ifdef HAVE_ASYNC_LDS
        // ASYNCcnt-tracked DMA: 16B per lane per issue, no VGPR roundtrip.
        #pragma unroll
        for (int i = 0; i < (C_CH * KC / 4) / 256; ++i) {   // 16 issues
            int flat = tid + 256 * i;                       // float4 index
            int r    = flat >> 6;                           // 64 float4 / row
            int c4   = flat & 63;
            const float* g = slab + (size_t)r * D_DIM + kc * KC + c4 * 4;
            float*       l = xs + r * PITCH + c4 * 4;
            __builtin_amdgcn_global_load_async_to_lds_b128(
                (v4i_t*)g, (AS3 v4i_t*)l, 0, 0);
        }
#if __has_builtin(__builtin_amdgcn_s_wait_asynccnt)
        __builtin_amdgcn_s_wait_asynccnt(0);
#else
        asm volatile("s_wait_asynccnt 0x0" ::: "memory");
#endif
        __syncthreads();
#else
        // prefetch next chunk (hint only; lowers to global_prefetch_b8)
        if (kc + 1 < D_DIM / KC) {
            const float* pf = slab + (size_t)(tid >> 2) * D_DIM
                                   + (kc + 1) * KC + (tid & 3) * 64;
            __builtin_prefetch(pf, 0, 1);
        }
        // cooperative load of 64 x 256 chunk, float4 granularity, coalesced
        const float4* src4 = (const float4*)(slab + (size_t)kc * KC);
        #pragma unroll
        for (int i = 0; i < (C_CH * KC / 4) / 256; ++i) {   // 16 iterations
            int flat = tid + 256 * i;                       // float4 index
            int r    = flat >> 6;                           // 64 float4 / row
            int c4   = flat & 63;
            float4 v = src4[(size_t)r * (D_DIM / 4) + c4];
            float* dst = xs + r * PITCH + c4 * 4;
            dst[0] = v.x; dst[1] = v.y; dst[2] = v.z; dst[3] = v.w;
        }
        __syncthreads();
#endif

        // (a) pooled-LN partial: thread t owns column kc*KC+t for all 64 rows
        {
            float accp = 0.f;
            #pragma unroll 4
            for (int r = 0; r < C_CH; ++r)
                accp += (xs[r * PITCH + tid] - mrow[r]) * rrow[r];
            const int d = kc * KC + tid;
            pooled_part[(size_t)(b * N_SP + n) * D_DIM + d] =
                gamma_pool[d] * accp + (float)C_CH * beta_pool[d];
        }

        // (b) covariance: V_WMMA_F32_16X16X4_F32 over this chunk.
        // A(16x4) and B(4x16) fp32 layouts share identical per-lane addressing:
        //   element = xs[(chBase + ln) * PITCH + k0 + 2*hi + {0,1}]
        // PITCH=260 -> dword bank = (4*ch + k) mod 64: conflict-free b64 loads.
        for (int k0 = 0; k0 < KC; k0 += 4) {
            const int kk = k0 + 2 * hi;
            v2f a0 = *(const v2f*)&xs[(ti0 + ln) * PITCH + kk];
            v2f b0 = *(const v2f*)&xs[(tj0 + ln) * PITCH + kk];
            acc0 = __builtin_amdgcn_wmma_f32_16x16x4_f32(
                       false, a0, false, b0, (short)0, acc0, false, false);
            v2f a1 = *(const v2f*)&xs[(ti1 + ln) * PITCH + kk];
            v2f b1 = *(const v2f*)&xs[(tj1 + ln) * PITCH + kk];
            acc1 = __builtin_amdgcn_wmma_f32_16x16x4_f32(
                       false, a1, false, b1, (short)0, acc1, false, false);
        }
        __syncthreads();
    }

    // store raw cov partials; C/D layout: VGPR g, lane<16 -> M=g, else M=g+8
    float* cp = cov_part + (size_t)(b * N_SP + n) * (C_CH * C_CH);
    #pragma unroll
    for (int g = 0; g < 8; ++g) {
        cp[(ti0 + g + 8 * hi) * C_CH + (tj0 + ln)] = acc0[g];
        cp[(ti1 + g + 8 * hi) * C_CH + (tj1 + ln)] = acc1[g];
    }
}

// ---------------------------------------------------------------------------
// Kernel 2: pooled = (1/L) * sum_n pooled_part  -> combined[:, 0:1024]
// ---------------------------------------------------------------------------
__global__ __launch_bounds__(256)
void reduce_pooled_kernel(const float* __restrict__ pooled_part,
                          float* __restrict__ combined)
{
    const int idx = blockIdx.x * 256 + threadIdx.x;   // b*D + d
    const int b = idx >> 10, d = idx & 1023;
    float s = 0.f;
    for (int n = 0; n < N_SP; ++n)
        s += pooled_part[(size_t)(b * N_SP + n) * D_DIM + d];
    combined[(size_t)b * COMB + d] = s * (1.0f / L_TOK);
}

// ---------------------------------------------------------------------------
// Kernel 3: cov = sum_n cov_part / (N*D) + eps*I
// ---------------------------------------------------------------------------
__global__ __launch_bounds__(256)
void reduce_cov_kernel(const float* __restrict__ cov_part,
                       float* __restrict__ cov)
{
    const int idx = blockIdx.x * 256 + threadIdx.x;   // b*4096 + ij
    const int b = idx >> 12, ij = idx & 4095;
    const int i = ij >> 6, j = ij & 63;
    float s = 0.f;
    for (int n = 0; n < N_SP; ++n)
        s += cov_part[(size_t)(b * N_SP + n) * 4096 + ij];
    float c = s * (1.0f / (N_SP * D_DIM));
    if (i == j) c += 1e-5f;
    cov[idx] = c;
}

// ---------------------------------------------------------------------------
// Kernel 4: per batch: Gauss-Jordan solve (cov+I) X = 2(cov-I)  (SPD, no
// pivoting needed), symmetrize, triu extract, layernorm(2080) -> combined tail
// ---------------------------------------------------------------------------
__global__ __launch_bounds__(256)
void solve_tangent_kernel(const float* __restrict__ cov,
                          const float* __restrict__ gamma_tan,
                          const float* __restrict__ beta_tan,
                          float* __restrict__ combined)
{
    __shared__ float As[C_CH * 66];
    __shared__ float Bs[C_CH * 66];
    __shared__ float fcol[C_CH];
    __shared__ float tanv[TRI];
    __shared__ float red1[8], red2[8];

    const int b   = blockIdx.x;
    const int tid = threadIdx.x;
    const float* cv = cov + (size_t)b * 4096;

    for (int idx = tid; idx < 4096; idx += 256) {
        int i = idx >> 6, j = idx & 63;
        float c = cv[idx];
        float e = (i == j) ? 1.0f : 0.0f;
        As[i * 66 + j] = c + e;
        Bs[i * 66 + j] = 2.0f * (c - e);
    }
    __syncthreads();

    for (int p = 0; p < C_CH; ++p) {
        float inv = 1.0f / As[p * 66 + p];
        __syncthreads();
        if (tid < 64)        As[p * 66 + tid]        *= inv;
        else if (tid < 128)  Bs[p * 66 + (tid - 64)] *= inv;
        __syncthreads();
        if (tid < 64) fcol[tid] = As[tid * 66 + p];
        __syncthreads();
        #pragma unroll 4
        for (int it = 0; it < 32; ++it) {            // 64 rows x 128 cols
            int idx = tid + 256 * it;
            int r = idx >> 7, c = idx & 127;
            if (r != p) {
                float f = fcol[r];
                if (c < 64) As[r * 66 + c]        -= f * As[p * 66 + c];
                else        Bs[r * 66 + (c - 64)] -= f * Bs[p * 66 + (c - 64)];
            }
        }
        __syncthreads();
    }

    // log_cov = 0.5*(Lm + Lm^T); row-major upper triangle
    for (int idx = tid; idx < 4096; idx += 256) {
        int i = idx >> 6, j = idx & 63;
        if (j >= i) {
            int t = i * C_CH - (i * (i - 1)) / 2 + (j - i);
            tanv[t] = 0.5f * (Bs[i * 66 + j] + Bs[j * 66 + i]);
        }
    }
    __syncthreads();

    // layernorm over the 2080 tangent entries
    float s1 = 0.f, s2 = 0.f;
    for (int t = tid; t < TRI; t += 256) { float v = tanv[t]; s1 += v; s2 += v * v; }
    for (int off = 16; off > 0; off >>= 1) {
        s1 += __shfl_xor(s1, off, 32);
        s2 += __shfl_xor(s2, off, 32);
    }
    if ((tid & 31) == 0) { red1[tid >> 5] = s1; red2[tid >> 5] = s2; }
    __syncthreads();
    s1 = 0.f; s2 = 0.f;
    for (int w = 0; w < 8; ++w) { s1 += red1[w]; s2 += red2[w]; }
    const float m    = s1 * (1.0f / TRI);
    const float var  = s2 * (1.0f / TRI) - m * m;
    const float rstd = rsqrtf(var + 1e-5f);

    float* outb = combined + (size_t)b * COMB + D_DIM;
    for (int t = tid; t < TRI; t += 256)
        outb[t] = (tanv[t] - m) * rstd * gamma_tan[t] + beta_tan[t];
}

// ---------------------------------------------------------------------------
// Kernel 5: out[b,k] = combined[b,:] . W[k,:] + bias[k]   (one wave / output)
// ---------------------------------------------------------------------------
__global__ __launch_bounds__(128)
void final_linear_kernel(const float* __restrict__ combined,
                         const float* __restrict__ W,
                         const float* __restrict__ bias,
                         float* __restrict__ out)
{
    const int wid  = (blockIdx.x * 128 + threadIdx.x) >> 5;  // exact: 1280 waves
    const int lane = threadIdx.x & 31;
    const int b = wid / K_OUT, k = wid % K_OUT;
    const float* cb = combined + (size_t)b * COMB;
    const float* wr = W        + (size_t)k * COMB;
    float s = 0.f;
    for (int j = lane; j < COMB; j += 32) s += cb[j] * wr[j];
    for (int off = 16; off > 0; off >>= 1) s += __shfl_xor(s, off, 32);
    if (lane == 0) out[b * K_OUT + k] = s + bias[k];
}

// ---------------------------------------------------------------------------
extern "C" void kernel_launch(void* const* d_in, const int* in_sizes, int n_in,
                              void* d_out, int out_size, void* d_ws, size_t ws_size,
                              hipStream_t stream)
{
    (void)in_sizes; (void)n_in; (void)out_size; (void)ws_size;
    const float* x          = (const float*)d_in[0];
    const float* gamma_pool = (const float*)d_in[1];
    const float* beta_pool  = (const float*)d_in[2];
    const float* gamma_tan  = (const float*)d_in[3];
    const float* beta_tan   = (const float*)d_in[4];
    const float* W_final    = (const float*)d_in[5];
    const float* b_final    = (const float*)d_in[6];
    float* out = (float*)d_out;

    float* ws          = (float*)d_ws;
    float* pooled_part = ws;                                           // 1.0M f
    float* cov_part    = pooled_part + (size_t)B_BATCH * N_SP * D_DIM; // 4.2M f
    float* cov         = cov_part    + (size_t)B_BATCH * N_SP * 4096;  // 131K f
    float* combined    = cov         + (size_t)B_BATCH * 4096;         //  99K f

    const size_t lds1 = (size_t)(C_CH * PITCH + 2 * C_CH) * sizeof(float);
    dim3 g1(N_SP, B_BATCH);
    fused_ln_cov_kernel<<<g1, 256, lds1, stream>>>(x, gamma_pool, beta_pool,
                                                   pooled_part, cov_part);
    reduce_pooled_kernel<<<(B_BATCH * D_DIM) / 256, 256, 0, stream>>>(pooled_part, combined);
    reduce_cov_kernel<<<(B_BATCH * 4096) / 256, 256, 0, stream>>>(cov_part, cov);
    solve_tangent_kernel<<<B_BATCH, 256, 0, stream>>>(cov, gamma_tan, beta_tan, combined);
    final_linear_kernel<<<(B_BATCH * K_OUT * 32) / 128, 128, 0, stream>>>(combined, W_final, b_final, out);
}